// Score_net_6158983102598
// MI455X (gfx1250) — compile-verified
//
#include <hip/hip_runtime.h>
#include <hip/hip_bf16.h>

typedef __attribute__((ext_vector_type(2))) float v2f;
typedef __attribute__((ext_vector_type(8))) float v8f;

#define B_MAX   64
#define CH      2048
#define FM      14
#define SPATIAL (FM * FM)      // 196
#define NWIN    837
#define NPROP   7

// ---------------- Kernel 1: channel reduction via WMMA f32 16x16x4 ----------
// xsum[b][s] = sum_c x[b][c][s].  B matrix = ones => D[m][n] = sum_k A[m][k].
// A layout (16x4 f32): lane L(0-15): M=L, K=0 (v0), K=1 (v1); lane L+16: K=2,3.
// Each wave reduces a 512-channel chunk for 16 spatial positions; 4 waves/block
// combine through LDS.
__global__ void __launch_bounds__(128)
sum_channels_wmma(const float* __restrict__ x, float* __restrict__ xsum) {
    const int tile = blockIdx.x;          // 0..12 (13 tiles of 16 spatial pos)
    const int b    = blockIdx.y;
    const int lane = threadIdx.x & 31;
    const int wave = threadIdx.x >> 5;    // 0..3

    int s = tile * 16 + (lane & 15);
    if (s > SPATIAL - 1) s = SPATIAL - 1;            // clamp: rows 4..15 of the
                                                     // last tile are discarded
    const int chalf = lane >> 4;                     // 0 or 1
    const float* p = x + ((size_t)b * CH + (size_t)(wave * 512 + 2 * chalf)) * SPATIAL + s;

    v2f ones = {1.0f, 1.0f};
    v8f acc = {};
    #pragma unroll 4
    for (int it = 0; it < 128; ++it) {               // 128 * 4 = 512 channels
        v2f a;
        a.x = p[0];                                  // channel c0 + 2*chalf
        a.y = p[SPATIAL];                            // channel c0 + 2*chalf + 1
        acc = __builtin_amdgcn_wmma_f32_16x16x4_f32(
            /*neg_a=*/false, a, /*neg_b=*/false, ones,
            /*c_mod=*/(short)0, acc, /*reuse_a=*/false, /*reuse_b=*/false);
        p += 4 * SPATIAL;
    }

    // D layout: lanes 0-15 hold column N=lane; VGPR r -> row M=r (lanes 0-15)
    // and M=r+8 (lanes 16-31). All columns identical (B = ones).
    __shared__ float part[4][16];
    if (lane == 0) {
        #pragma unroll
        for (int r = 0; r < 8; ++r) part[wave][r] = acc[r];
    } else if (lane == 16) {
        #pragma unroll
        for (int r = 0; r < 8; ++r) part[wave][8 + r] = acc[r];
    }
    __syncthreads();
    const int tid = threadIdx.x;
    if (tid < 16) {
        int sp = tile * 16 + tid;
        if (sp < SPATIAL) {
            float v = part[0][tid] + part[1][tid] + part[2][tid] + part[3][tid];
            xsum[(size_t)b * SPATIAL + sp] = v;
        }
    }
}

// ---------------- Kernel 2: valid avg-pool window scores --------------------
__constant__ int c_rh[11]  = {4, 3, 5, 6, 5, 7, 8, 6, 10, 7, 9};
__constant__ int c_rw[11]  = {4, 5, 3, 6, 7, 5, 8, 10, 6, 9, 7};
__constant__ int c_off[12] = {0, 121, 241, 361, 442, 522, 602, 651, 696, 741, 789, 837};

__global__ void __launch_bounds__(256)
window_scores_kernel(const float* __restrict__ xsum, float* __restrict__ wsc) {
    const int b = blockIdx.x;
    __shared__ float sh[SPATIAL];
    for (int i = threadIdx.x; i < SPATIAL; i += blockDim.x)
        sh[i] = xsum[(size_t)b * SPATIAL + i];
    __syncthreads();

    for (int w = threadIdx.x; w < NWIN; w += blockDim.x) {
        int r = 0;
        while (w >= c_off[r + 1]) ++r;
        const int local = w - c_off[r];
        const int rh = c_rh[r], rw = c_rw[r];
        const int ow = FM - rw + 1;
        const int i0 = local / ow, j0 = local % ow;
        float sum = 0.0f;
        for (int di = 0; di < rh; ++di)
            for (int dj = 0; dj < rw; ++dj)
                sum += sh[(i0 + di) * FM + (j0 + dj)];
        wsc[(size_t)b * NWIN + w] = sum / (float)(rh * rw);
    }
}

// ---------------- Kernel 3: greedy IoU-NMS, one wave per (group,batch) ------
// Faithful to reference: +1 box lengths, keep iou <= thresh, argmax with
// first-index tie-break, output score = original window score at the index.
__global__ void __launch_bounds__(32)
nms_kernel(const float* __restrict__ wsc, const float* __restrict__ coords,
           float* __restrict__ out_idx, float* __restrict__ out_sc) {
    const int g    = blockIdx.x;   // 0..2
    const int b    = blockIdx.y;
    const int lane = threadIdx.x;  // wave32

    const int start = (g == 0) ? 0 : (g == 1) ? 361 : 602;
    const int gsize = (g == 0) ? 361 : (g == 1) ? 241 : 235;
    const int kSel  = (g == 0) ? 2 : (g == 1) ? 3 : 2;
    const int ooff  = (g == 0) ? 0 : (g == 1) ? 2 : 5;
    const float thresh = 0.25f;
    const float NEG = -__builtin_inff();

    // Each lane owns elements lane, lane+32, ... (<= 12 for gsize <= 361).
    float sc[12], bx0[12], by0[12], bx1[12], by1[12], ar[12];
    #pragma unroll
    for (int t = 0; t < 12; ++t) {
        const int gg = lane + 32 * t;
        if (gg < gsize) {
            sc[t] = wsc[(size_t)b * NWIN + start + gg];
            const float* c = coords + (size_t)(start + gg) * 4;
            bx0[t] = c[0]; by0[t] = c[1]; bx1[t] = c[2]; by1[t] = c[3];
            ar[t] = (bx1[t] - bx0[t] + 1.0f) * (by1[t] - by0[t] + 1.0f);
        } else {
            sc[t] = NEG; bx0[t] = by0[t] = bx1[t] = by1[t] = 0.0f; ar[t] = 1.0f;
        }
    }

    for (int step = 0; step < kSel; ++step) {
        // local argmax (smaller index wins ties, matching jnp.argmax)
        float bs = NEG; int bi = 0x7fffffff;
        #pragma unroll
        for (int t = 0; t < 12; ++t) {
            const int gg = lane + 32 * t;
            if (gg < gsize && (sc[t] > bs || (sc[t] == bs && gg < bi))) {
                bs = sc[t]; bi = gg;
            }
        }
        // wave-wide argmax reduction
        for (int m = 16; m > 0; m >>= 1) {
            const float os = __shfl_xor(bs, m, 32);
            const int   oi = __shfl_xor(bi, m, 32);
            if (os > bs || (os == bs && oi < bi)) { bs = os; bi = oi; }
        }

        const float* cb = coords + (size_t)(start + bi) * 4;
        const float Bx0 = cb[0], By0 = cb[1], Bx1 = cb[2], By1 = cb[3];
        const float Bar = (Bx1 - Bx0 + 1.0f) * (By1 - By0 + 1.0f);

        if (lane == 0) {
            out_idx[(size_t)b * NPROP + ooff + step] = (float)(start + bi);
            out_sc [(size_t)b * NPROP + ooff + step] =
                wsc[(size_t)b * NWIN + start + bi];   // take_along_axis semantics
        }

        // suppress: chosen index + everything with iou > thresh
        #pragma unroll
        for (int t = 0; t < 12; ++t) {
            const int gg = lane + 32 * t;
            if (gg < gsize) {
                const float lt0 = fmaxf(bx0[t], Bx0), lt1 = fmaxf(by0[t], By0);
                const float rb0 = fminf(bx1[t], Bx1), rb1 = fminf(by1[t], By1);
                const float ww = rb0 - lt0 + 1.0f, hh = rb1 - lt1 + 1.0f;
                const float inter = (ww > 0.0f && hh > 0.0f) ? (ww * hh) : 0.0f;
                const float iou = inter / (ar[t] + Bar - inter);
                if (iou > thresh || gg == bi) sc[t] = NEG;
            }
        }
    }
}

// ---------------- launch ----------------------------------------------------
extern "C" void kernel_launch(void* const* d_in, const int* in_sizes, int n_in,
                              void* d_out, int out_size, void* d_ws, size_t ws_size,
                              hipStream_t stream) {
    const float* x      = (const float*)d_in[0];
    const float* coords = (const float*)d_in[1];
    (void)n_in; (void)out_size; (void)ws_size;

    const int B = in_sizes[0] / (CH * SPATIAL);      // 64

    float* out     = (float*)d_out;
    float* out_idx = out;                            // [B,7] indices (as float)
    float* out_sc  = out + (size_t)B * NPROP;        // [B,7] scores
    float* wsc     = out + (size_t)2 * B * NPROP;    // [B,837] window_scores
    float* xsum    = (float*)d_ws;                   // [B,196] scratch

    dim3 g1((SPATIAL + 15) / 16, B);                 // 13 x B
    sum_channels_wmma<<<g1, 128, 0, stream>>>(x, xsum);

    window_scores_kernel<<<B, 256, 0, stream>>>(xsum, wsc);

    dim3 g3(3, B);
    nms_kernel<<<g3, 32, 0, stream>>>(wsc, coords, out_idx, out_sc);
}